// SimplifiedMomentHead_79413945303748
// MI455X (gfx1250) — compile-verified
//
#include <hip/hip_runtime.h>
#include <hip/hip_bf16.h>

// SimplifiedMomentHead on gfx1250 (MI455X), fp32 end-to-end using
// V_WMMA_F32_16X16X4_F32 for every GEMM stage, double-buffered LDS
// pipelines, global_prefetch, and GLOBAL_LOAD_ASYNC_TO_LDS_B128 streaming
// of tiles (builtin confirmed present; takes int4-vector pointers).
//
// Shapes: B=64, N=256, D=384, DOUT=512, SID=73920.

typedef __attribute__((ext_vector_type(2))) float v2f;
typedef __attribute__((ext_vector_type(8))) float v8f;

constexpr int BN   = 64;     // batches
constexpr int NN   = 256;    // tokens
constexpr int DN   = 384;    // d_in
constexpr int DO   = 512;    // d_out
constexpr int SID  = DN * (DN + 1) / 2;  // 73920
constexpr float EPS = 1e-5f;

// --- CDNA5 async global->LDS path (guarded: falls back cleanly) ---
#if defined(__HIP_DEVICE_COMPILE__) && defined(__gfx1250__) && \
    __has_builtin(__builtin_amdgcn_global_load_async_to_lds_b128)
#define ASYNC_LDS 1
typedef int v4i __attribute__((vector_size(16)));
typedef __attribute__((address_space(1))) v4i* gbl_v4;  // global int4*
typedef __attribute__((address_space(3))) v4i* lds_v4;  // LDS int4*
#else
#define ASYNC_LDS 0
#endif

__device__ __forceinline__ void wait_async_zero() {
#if ASYNC_LDS
#if __has_builtin(__builtin_amdgcn_s_wait_asynccnt)
  __builtin_amdgcn_s_wait_asynccnt(0);
#else
  asm volatile("s_wait_asynccnt 0x0" ::: "memory");
#endif
#endif
}

// ---------------------------------------------------------------------------
// Tiled WMMA fp32 GEMM: C(M x N) = A(M x K) @ B(K x N), 64x64 tile per block,
// 256 threads = 8 waves; each wave owns two 16x16 WMMA accumulators.
// Double-buffered LDS: next K-panel is fetched (async-to-LDS when available,
// else reg-staged) while the current panel feeds the WMMAs.
// TRANSA=1 -> A supplied physically as (K x M) row-major (z_c^T case).
// MODE: 0 store, 1 store (3I - P), 2 store smul*P, 3 atomicAdd(P).
// split_k: blockIdx.z selects K-chunk (batch 0) instead of batch.
// ---------------------------------------------------------------------------
template <int TRANSA, int MODE>
__global__ __launch_bounds__(256) void gemm64_wmma(
    const float* __restrict__ A, const float* __restrict__ B,
    float* __restrict__ C, int M, int N, int K, int lda, int ldb, int ldc,
    long sA, long sB, long sC, int split_k, int kchunk,
    const float* __restrict__ scale_ptr, float scale_const) {
  // +4-float padding: rows stay 16B-aligned (B128 async legal) and the
  // 16-lane fragment reads still map to 16 distinct LDS banks.
  __shared__ float As[2][64][20];  // [buf][m][k]
  __shared__ float Bs[2][16][68];  // [buf][k][n]

  const int tid  = threadIdx.x;
  const int w    = tid >> 5;
  const int lane = tid & 31;
  const int l15  = lane & 15;
  const int hi   = (lane >> 4) & 1;

  const int bz = blockIdx.z;
  const float* Ab;
  const float* Bb;
  float* Cb;
  int kbeg, kend, batch;
  if (split_k) {
    Ab = A; Bb = B; Cb = C;
    kbeg = bz * kchunk;
    kend = kbeg + kchunk; if (kend > K) kend = K;
    batch = 0;
  } else {
    Ab = A + (long)bz * sA; Bb = B + (long)bz * sB; Cb = C + (long)bz * sC;
    kbeg = 0; kend = K; batch = bz;
  }

  const int mbase = blockIdx.y * 64;
  const int nbase = blockIdx.x * 64;
  const int waveM = w >> 2;   // 0..1
  const int waveN = w & 3;    // 0..3

  // per-thread tile-staging roles
  const int a_lr = tid >> 2;          // TRANSA==0: row 0..63
  const int a_lk = (tid & 3) << 2;    // TRANSA==0: k 0,4,8,12
  const int a_pk = tid >> 4;          // TRANSA==1: k 0..15
  const int a_pm = (tid & 15) << 2;   // TRANSA==1: m 0..60
  const int b_pk = tid >> 4;          // k 0..15
  const int b_pn = (tid & 15) << 2;   // n 0..60

  auto aGlobal = [&](int kk) -> const float* {
    if (TRANSA == 0) return Ab + (long)(mbase + a_lr) * lda + kk + a_lk;
    return Ab + (long)(kk + a_pk) * lda + mbase + a_pm;
  };
  auto bGlobal = [&](int kk) -> const float* {
    return Bb + (long)(kk + b_pk) * ldb + nbase + b_pn;
  };
  auto stageA = [&](const float4& va, int buf) {
    if (TRANSA == 0) {
      As[buf][a_lr][a_lk + 0] = va.x; As[buf][a_lr][a_lk + 1] = va.y;
      As[buf][a_lr][a_lk + 2] = va.z; As[buf][a_lr][a_lk + 3] = va.w;
    } else {
      As[buf][a_pm + 0][a_pk] = va.x; As[buf][a_pm + 1][a_pk] = va.y;
      As[buf][a_pm + 2][a_pk] = va.z; As[buf][a_pm + 3][a_pk] = va.w;
    }
  };
  auto stageB = [&](const float4& vb, int buf) {
    Bs[buf][b_pk][b_pn + 0] = vb.x; Bs[buf][b_pk][b_pn + 1] = vb.y;
    Bs[buf][b_pk][b_pn + 2] = vb.z; Bs[buf][b_pk][b_pn + 3] = vb.w;
  };

  v8f acc0 = {};
  v8f acc1 = {};
  float4 va_next, vb_next;
  (void)vb_next;

  // ---- prologue: stage first K-panel into buffer 0 ----
#if ASYNC_LDS
  if (TRANSA == 0) {
    __builtin_amdgcn_global_load_async_to_lds_b128(
        (gbl_v4)aGlobal(kbeg), (lds_v4)&As[0][a_lr][a_lk], 0, 0);
  } else {
    va_next = *(const float4*)aGlobal(kbeg);
    stageA(va_next, 0);
  }
  __builtin_amdgcn_global_load_async_to_lds_b128(
      (gbl_v4)bGlobal(kbeg), (lds_v4)&Bs[0][b_pk][b_pn], 0, 0);
  wait_async_zero();
#else
  va_next = *(const float4*)aGlobal(kbeg);
  vb_next = *(const float4*)bGlobal(kbeg);
  stageA(va_next, 0);
  stageB(vb_next, 0);
#endif
  __syncthreads();

  int cur = 0;
  for (int kk = kbeg; kk < kend; kk += 16) {
    const bool have = (kk + 16) < kend;
    const int nxt = cur ^ 1;

    // ---- issue next panel's fetches before computing ----
    if (have) {
#if ASYNC_LDS
      if (TRANSA == 0) {
        __builtin_amdgcn_global_load_async_to_lds_b128(
            (gbl_v4)aGlobal(kk + 16), (lds_v4)&As[nxt][a_lr][a_lk], 0, 0);
      } else {
        va_next = *(const float4*)aGlobal(kk + 16);
      }
      __builtin_amdgcn_global_load_async_to_lds_b128(
          (gbl_v4)bGlobal(kk + 16), (lds_v4)&Bs[nxt][b_pk][b_pn], 0, 0);
#else
      va_next = *(const float4*)aGlobal(kk + 16);
      vb_next = *(const float4*)bGlobal(kk + 16);
#endif
      if (kk + 32 < kend) {  // L2 prefetch two panels ahead
        __builtin_prefetch(aGlobal(kk + 32), 0, 1);
        __builtin_prefetch(bGlobal(kk + 32), 0, 1);
      }
    }

    // ---- compute on buffer `cur` ----
    const int rm0 = waveM * 32;
    const int rm1 = waveM * 32 + 16;
    const int cn  = waveN * 16;
#pragma unroll
    for (int k4 = 0; k4 < 4; ++k4) {
      const int kb = k4 * 4;
      // 16x4 fp32 A fragment: lanes 0-15 hold K=0/1, lanes 16-31 hold K=2/3
      v2f a0, a1, bf;
      a0.x = As[cur][rm0 + l15][kb + (hi ? 2 : 0)];
      a0.y = As[cur][rm0 + l15][kb + (hi ? 3 : 1)];
      a1.x = As[cur][rm1 + l15][kb + (hi ? 2 : 0)];
      a1.y = As[cur][rm1 + l15][kb + (hi ? 3 : 1)];
      bf.x = Bs[cur][kb + (hi ? 2 : 0)][cn + l15];
      bf.y = Bs[cur][kb + (hi ? 3 : 1)][cn + l15];
      acc0 = __builtin_amdgcn_wmma_f32_16x16x4_f32(false, a0, false, bf,
                                                   (short)0, acc0, false, false);
      acc1 = __builtin_amdgcn_wmma_f32_16x16x4_f32(false, a1, false, bf,
                                                   (short)0, acc1, false, false);
    }

    // ---- commit next panel ----
    if (have) {
#if ASYNC_LDS
      if (TRANSA != 0) stageA(va_next, nxt);
      wait_async_zero();
#else
      stageA(va_next, nxt);
      stageB(vb_next, nxt);
#endif
    }
    __syncthreads();
    cur = nxt;
  }

  // ---- epilogue ----
  float smul = 1.0f;
  if (MODE == 2) smul = scale_const * (scale_ptr ? scale_ptr[batch] : 1.0f);
  const int col = nbase + waveN * 16 + l15;
#pragma unroll
  for (int r = 0; r < 8; ++r) {
    const int row0 = mbase + waveM * 32 + r + (hi ? 8 : 0);
    const int row1 = row0 + 16;
    const float v0 = acc0[r];
    const float v1 = acc1[r];
    if (MODE == 0) {
      Cb[(long)row0 * ldc + col] = v0;
      Cb[(long)row1 * ldc + col] = v1;
    } else if (MODE == 1) {
      Cb[(long)row0 * ldc + col] = (row0 == col ? 3.0f : 0.0f) - v0;
      Cb[(long)row1 * ldc + col] = (row1 == col ? 3.0f : 0.0f) - v1;
    } else if (MODE == 2) {
      Cb[(long)row0 * ldc + col] = v0 * smul;
      Cb[(long)row1 * ldc + col] = v1 * smul;
    } else {
      atomicAdd(&Cb[(long)row0 * ldc + col], v0);
      atomicAdd(&Cb[(long)row1 * ldc + col], v1);
    }
  }
}

// ---------------------------------------------------------------------------
// Elementwise / reduction helpers
// ---------------------------------------------------------------------------
__global__ __launch_bounds__(256) void row_inv_deg_k(
    const float* __restrict__ g, float* __restrict__ inv_deg) {
  const int row = blockIdx.x;  // b*N + m
  __shared__ float s[256];
  s[threadIdx.x] = g[(long)row * NN + threadIdx.x];
  __syncthreads();
  for (int o = 128; o > 0; o >>= 1) {
    if (threadIdx.x < o) s[threadIdx.x] += s[threadIdx.x + o];
    __syncthreads();
  }
  if (threadIdx.x == 0) inv_deg[row] = 1.0f / (s[0] + EPS);
}

__global__ __launch_bounds__(256) void weights_k(
    const float* __restrict__ g, const float* __restrict__ inv_deg,
    float* __restrict__ W) {
  const long i = (long)blockIdx.x * 256 + threadIdx.x;  // b*N*N + m*N + n
  W[i] = g[i] * inv_deg[i >> 8];
}

__global__ __launch_bounds__(256) void colw_k(const float* __restrict__ W,
                                              float* __restrict__ colw) {
  const int b = blockIdx.x, n = threadIdx.x;
  float s = 0.0f;
  const float* Wb = W + (long)b * NN * NN;
  for (int m = 0; m < NN; ++m) s += Wb[(long)m * NN + n];
  colw[b * NN + n] = s;
}

__global__ __launch_bounds__(384) void wmean_zc_k(
    const float* __restrict__ tokens, const float* __restrict__ colw,
    float* __restrict__ zc) {
  const int b = blockIdx.x, d = threadIdx.x;
  const float* tb = tokens + (long)b * NN * DN;
  float* zb = zc + (long)b * NN * DN;
  float wm = 0.0f;
  for (int n = 0; n < NN; ++n) wm += colw[b * NN + n] * tb[(long)n * DN + d];
  for (int n = 0; n < NN; ++n) zb[(long)n * DN + d] = tb[(long)n * DN + d] - wm;
}

__global__ __launch_bounds__(384) void trace_k(const float* __restrict__ M2,
                                               float* __restrict__ invTr,
                                               float* __restrict__ sc2) {
  const int b = blockIdx.x, d = threadIdx.x;
  __shared__ float s[384];
  s[d] = M2[(long)b * DN * DN + (long)d * DN + d];
  __syncthreads();
  if (d < 128) s[d] += s[d + 256];
  __syncthreads();
  for (int o = 128; o > 0; o >>= 1) {
    if (d < o) s[d] += s[d + o];
    __syncthreads();
  }
  if (d == 0) {
    const float tr = s[0] + EPS;
    invTr[b] = 1.0f / tr;
    sc2[b]   = 0.5f * rsqrtf(tr);
  }
}

__global__ __launch_bounds__(256) void scale_mn_k(float* __restrict__ Mn,
                                                  const float* __restrict__ invTr) {
  const long i = (long)blockIdx.x * 256 + threadIdx.x;
  Mn[i] *= invTr[i / (DN * DN)];
}

__global__ __launch_bounds__(256) void init_y1_k(const float* __restrict__ Mn,
                                                 float* __restrict__ Y) {
  const long i = (long)blockIdx.x * 256 + threadIdx.x;
  const int r = (int)((i / DN) % DN);
  const int c = (int)(i % DN);
  Y[i] = (r == c ? 1.5f : 0.0f) - 0.5f * Mn[i];
}

__global__ __launch_bounds__(384) void pack_triu_k(const float* __restrict__ Y,
                                                   float* __restrict__ Atri) {
  const int i = blockIdx.x;   // row 0..383
  const int b = blockIdx.y;
  const int j = threadIdx.x;
  if (j >= i) {
    const int off = i * DN - (i * (i - 1)) / 2;  // start of row i in triu vec
    Atri[(long)b * SID + off + (j - i)] = Y[(long)b * DN * DN + (long)i * DN + j];
  }
}

__global__ __launch_bounds__(256) void zero_k(float* __restrict__ p) {
  p[(long)blockIdx.x * 256 + threadIdx.x] = 0.0f;
}

__global__ __launch_bounds__(256) void bias_gelu_k(const float* __restrict__ acc,
                                                   const float* __restrict__ b2,
                                                   float* __restrict__ out) {
  const long i = (long)blockIdx.x * 256 + threadIdx.x;  // b*DO + o
  const float x = acc[i] + b2[i & (DO - 1)];
  out[i] = 0.5f * x * (1.0f + erff(x * 0.70710678118654752f));
}

// ---------------------------------------------------------------------------
extern "C" void kernel_launch(void* const* d_in, const int* in_sizes, int n_in,
                              void* d_out, int out_size, void* d_ws,
                              size_t ws_size, hipStream_t stream) {
  const float* tokens = (const float*)d_in[0];  // (64,256,384)
  const float* graph  = (const float*)d_in[1];  // (64,256,256)
  const float* W2     = (const float*)d_in[2];  // (73920,512)
  const float* b2     = (const float*)d_in[3];  // (512,)
  float* out = (float*)d_out;                   // (64,512)
  float* ws  = (float*)d_ws;

  // workspace layout (floats)
  float* W      = ws;                                     // 64*256*256
  float* ZC     = W + (long)BN * NN * NN;                 // 64*256*384
  float* WZ     = ZC + (long)BN * NN * DN;                // 64*256*384
  float* MN     = WZ + (long)BN * NN * DN;                // 64*384*384
  float* YA     = MN + (long)BN * DN * DN;                // 64*384*384
  float* TT     = YA + (long)BN * DN * DN;                // 64*384*384
  float* invdeg = TT + (long)BN * DN * DN;                // 16384
  float* colw   = invdeg + BN * NN;                       // 16384
  float* invTr  = colw + BN * NN;                         // 64
  float* sc2    = invTr + BN;                             // 64
  float* acc    = sc2 + BN;                               // 64*512
  float* YB     = ZC;  // alias: z_c/wz dead after M2 GEMM (50.4MB >= 37.7MB)
  float* ATRI   = MN;  // alias: Mn dead after last T GEMM (37.7MB >= 18.9MB)

  // 1-4: graph normalization, column weights, centered tokens
  row_inv_deg_k<<<BN * NN, 256, 0, stream>>>(graph, invdeg);
  weights_k<<<(BN * NN * NN) / 256, 256, 0, stream>>>(graph, invdeg, W);
  colw_k<<<BN, 256, 0, stream>>>(W, colw);
  wmean_zc_k<<<BN, 384, 0, stream>>>(tokens, colw, ZC);

  // 5: wz = weights @ z_c  (256x256x384 per batch)
  gemm64_wmma<0, 0><<<dim3(DN / 64, NN / 64, BN), 256, 0, stream>>>(
      W, ZC, WZ, NN, DN, NN, NN, DN, DN, (long)NN * NN, (long)NN * DN,
      (long)NN * DN, 0, 0, nullptr, 1.0f);

  // 6: M2 = z_c^T @ wz  (384x256x384 per batch; A physically K x M)
  gemm64_wmma<1, 0><<<dim3(DN / 64, DN / 64, BN), 256, 0, stream>>>(
      ZC, WZ, MN, DN, DN, NN, DN, DN, DN, (long)NN * DN, (long)NN * DN,
      (long)DN * DN, 0, 0, nullptr, 1.0f);

  // 7: trace, Mn = M2/(tr+eps)
  trace_k<<<BN, 384, 0, stream>>>(MN, invTr, sc2);
  scale_mn_k<<<(BN * DN * DN) / 256, 256, 0, stream>>>(MN, invTr);

  // 8: Y1 = 1.5I - 0.5*Mn  (NS iter 1, Y0 = I)
  init_y1_k<<<(BN * DN * DN) / 256, 256, 0, stream>>>(MN, YA);

  const dim3 gNS(DN / 64, DN / 64, BN);
  const long sD = (long)DN * DN;
  // 9a: iter 2:  T = 3I - Mn@Y1 ; Y2 = 0.5*Y1@T
  gemm64_wmma<0, 1><<<gNS, 256, 0, stream>>>(MN, YA, TT, DN, DN, DN, DN, DN, DN,
                                             sD, sD, sD, 0, 0, nullptr, 1.0f);
  gemm64_wmma<0, 2><<<gNS, 256, 0, stream>>>(YA, TT, YB, DN, DN, DN, DN, DN, DN,
                                             sD, sD, sD, 0, 0, nullptr, 0.5f);
  // 9b: iter 3:  T = 3I - Mn@Y2 ; Y3 = (0.5*rsqrt(tr+eps)) * Y2@T
  gemm64_wmma<0, 1><<<gNS, 256, 0, stream>>>(MN, YB, TT, DN, DN, DN, DN, DN, DN,
                                             sD, sD, sD, 0, 0, nullptr, 1.0f);
  gemm64_wmma<0, 2><<<gNS, 256, 0, stream>>>(YB, TT, YA, DN, DN, DN, DN, DN, DN,
                                             sD, sD, sD, 0, 0, sc2, 1.0f);

  // 10: pack upper triangle of Y3 into Atri (64 x 73920)
  pack_triu_k<<<dim3(DN, BN), 384, 0, stream>>>(YA, ATRI);

  // 11: acc = Atri @ W2, split-K=60 (chunk 1232, multiple of 16), atomic f32
  zero_k<<<(BN * DO) / 256, 256, 0, stream>>>(acc);
  gemm64_wmma<0, 3><<<dim3(DO / 64, 1, 60), 256, 0, stream>>>(
      ATRI, W2, acc, BN, DO, SID, SID, DO, DO, 0, 0, 0, 1, SID / 60, nullptr,
      1.0f);

  // 12: out = gelu(acc + b2)
  bias_gelu_k<<<(BN * DO) / 256, 256, 0, stream>>>(acc, b2, out);
}